// ATFA_89859305767672
// MI455X (gfx1250) — compile-verified
//
#include <hip/hip_runtime.h>

// ---------------------------------------------------------------------------
// MI455X (gfx1250): dual-attention + conv block, fp32 via V_WMMA_F32_16X16X4_F32.
// Zero-page instead of predicated loads, async global->LDS staging of conv
// weights (GLOBAL_LOAD_ASYNC_TO_LDS_B128), 2x M-tiling, and register
// double-buffering of LDS B-fragments to hide ds latency.
// ---------------------------------------------------------------------------

typedef float v2f __attribute__((ext_vector_type(2)));
typedef float v4f __attribute__((ext_vector_type(4)));
typedef float v8f __attribute__((ext_vector_type(8)));
typedef int   v4i_ __attribute__((vector_size(4 * sizeof(int))));
typedef __attribute__((address_space(1))) v4i_* gas_v4i;
typedef __attribute__((address_space(3))) v4i_* las_v4i;

#if __has_builtin(__builtin_amdgcn_global_load_async_to_lds_b128)
#define HAVE_ASYNC_LDS 1
#else
#define HAVE_ASYNC_LDS 0
#endif

__device__ __forceinline__ v8f wmma4(v2f a, v2f b, v8f c) {
  return __builtin_amdgcn_wmma_f32_16x16x4_f32(false, a, false, b, (short)0, c,
                                               false, false);
}

// 16-byte global -> LDS copy; async path when the toolchain exposes it.
__device__ __forceinline__ void copy16_g2l(const float* g, float* l) {
#if HAVE_ASYNC_LDS
  __builtin_amdgcn_global_load_async_to_lds_b128(
      (gas_v4i)const_cast<float*>(g), (las_v4i)l, 0, 0);
#else
  *(v4f*)l = *(const v4f*)g;
#endif
}

__device__ __forceinline__ void async_join() {
#if HAVE_ASYNC_LDS
#if __has_builtin(__builtin_amdgcn_s_wait_asynccnt)
  __builtin_amdgcn_s_wait_asynccnt(0);
#else
  asm volatile("s_wait_asynccnt 0" ::: "memory");
#endif
#endif
}

// Problem constants.
constexpr int B_ = 2, T_ = 249, F_ = 64, C_ = 256;
constexpr int KQK  = 9 * C_;     // 2304
constexpr int KFIN = 9 * 768;    // 6912
constexpr long NS = (long)B_ * F_ * T_ * C_;

// Workspace layout (floats).
constexpr long OFF_QF = 0;        // [B][F][T][C]
constexpr long OFF_KF = 1 * NS;
constexpr long OFF_VF = 2 * NS;
constexpr long OFF_QT = 3 * NS;   // [B][T][F][C]
constexpr long OFF_KT = 4 * NS;
constexpr long OFF_VT = 5 * NS;
constexpr long OFF_GF = 6 * NS;
constexpr long OFF_GT = 7 * NS;
constexpr long OFF_WB = 8 * NS;                  // 6x qkv weights N-major [256][2304]
constexpr long WSZ_QKV = (long)KQK * C_;
constexpr long OFF_WFIN = OFF_WB + 6 * WSZ_QKV;  // final weights N-major [64][6912]
constexpr long OFF_ZERO = OFF_WFIN + (long)KFIN * 64;  // 256-float zero page

__global__ void k_zero(float* __restrict__ p, int n) {
  int i = blockIdx.x * blockDim.x + threadIdx.x;
  if (i < n) p[i] = 0.f;
}

// ---------------------------------------------------------------------------
// Weight repacks to N-major [O][K].
// ---------------------------------------------------------------------------
__global__ void k_transpose(const float* __restrict__ w, float* __restrict__ bt,
                            int K, int N) {
  long i = (long)blockIdx.x * blockDim.x + threadIdx.x;
  if (i >= (long)K * N) return;
  int  n = (int)(i % N);
  long k = i / N;
  bt[(long)n * K + k] = w[i];
}

// Freq block convolves the (F,T)-transposed image: swap kh/kw while transposing.
__global__ void k_freq_repack(const float* __restrict__ w, float* __restrict__ bt) {
  long i = (long)blockIdx.x * blockDim.x + threadIdx.x;
  if (i >= (long)KQK * C_) return;
  int  o  = (int)(i % C_);
  long r  = i / C_;
  int  c  = (int)(r % C_);
  int  pp = (int)(r / C_);
  int  kh = pp / 3, kw = pp % 3;
  bt[(long)o * KQK + (kw * 3 + kh) * C_ + c] = w[i];
}

// ---------------------------------------------------------------------------
// Q/K/V convs: implicit-im2col GEMM, M=32 (2 m-subtiles/wave), N=256, K=2304.
// Weights staged in LDS via async copies; A via zero-page pointer select.
// ---------------------------------------------------------------------------
struct Bias6 { const float* p[6]; };

constexpr int CKQ = 32;  // K-chunk staged per barrier round

__global__ __launch_bounds__(128)
void k_qkv_conv(const float* __restrict__ x, float* __restrict__ ws, Bias6 biases) {
  __shared__ float bLds[256][36];  // [n][k-chunk], stride 36: 16B-aligned rows,
                                   // conflict-free lane pattern

  int id = blockIdx.x;
  int tensor = id % 6; id /= 6;
  int half   = id % 2; id /= 2;
  int t = id % T_;
  int b = id / T_;

  int wave = threadIdx.x >> 5, lane = threadIdx.x & 31;
  int l15 = lane & 15, hi = lane >> 4;

  const float* bt   = ws + OFF_WB + (long)tensor * WSZ_QKV;
  const float* zp   = ws + OFF_ZERO;
  const float* bias = biases.p[tensor];
  float*       outp = ws + (long)tensor * NS;

  v8f acc[2][4];
  #pragma unroll
  for (int mm = 0; mm < 2; ++mm)
    #pragma unroll
    for (int j = 0; j < 4; ++j)
      acc[mm][j] = (v8f){0.f,0.f,0.f,0.f,0.f,0.f,0.f,0.f};

  int f0 = half * 32;

  for (int p = 0; p < 9; ++p) {
    int dt = p / 3 - 1, df = p % 3 - 1;
    int tt = t + dt;
    bool tOK = (unsigned)tt < (unsigned)T_;
    const float* aPtr[2];
    #pragma unroll
    for (int mm = 0; mm < 2; ++mm) {
      int ff = f0 + mm * 16 + l15 + df;
      bool v = tOK && ((unsigned)ff < (unsigned)F_);
      aPtr[mm] = v ? x + (((long)b * T_ + tt) * F_ + ff) * C_ : zp;
    }
    for (int cb = 0; cb < C_ / CKQ; ++cb) {
      int kbase = p * C_ + cb * CKQ;
      __syncthreads();  // previous chunk fully consumed
      // Cooperative copy: 256 rows x 32 floats = 2048 x 16B, 16 per thread.
      #pragma unroll
      for (int i = 0; i < 16; ++i) {
        int e = threadIdx.x + i * 128;
        int n = e >> 3;
        int sub = e & 7;
        copy16_g2l(bt + (long)n * KQK + kbase + sub * 4, &bLds[n][sub * 4]);
      }
      async_join();
      __syncthreads();

      int cA = cb * CKQ;
      // Register double-buffer of B fragments: ds latency hidden behind the
      // previous step's 8 WMMAs.
      v2f bfrag[2][4];
      #pragma unroll
      for (int j = 0; j < 4; ++j)
        bfrag[0][j] = *(const v2f*)(&bLds[(wave * 4 + j) * 16 + l15][hi * 2]);
      #pragma unroll
      for (int s = 0; s < CKQ / 4; ++s) {
        int cur = s & 1;
        if (s + 1 < CKQ / 4) {
          #pragma unroll
          for (int j = 0; j < 4; ++j)
            bfrag[cur ^ 1][j] =
                *(const v2f*)(&bLds[(wave * 4 + j) * 16 + l15][(s + 1) * 4 + hi * 2]);
        }
        v2f a0 = *(const v2f*)(aPtr[0] + cA + s * 4 + hi * 2);
        v2f a1 = *(const v2f*)(aPtr[1] + cA + s * 4 + hi * 2);
        #pragma unroll
        for (int j = 0; j < 4; ++j) {
          acc[0][j] = wmma4(a0, bfrag[cur][j], acc[0][j]);
          acc[1][j] = wmma4(a1, bfrag[cur][j], acc[1][j]);
        }
      }
    }
  }

  #pragma unroll
  for (int mm = 0; mm < 2; ++mm) {
    #pragma unroll
    for (int j = 0; j < 4; ++j) {
      int n = (wave * 4 + j) * 16 + l15;
      float bv = bias[n];
      #pragma unroll
      for (int i = 0; i < 8; ++i) {
        int f = f0 + mm * 16 + i + hi * 8;
        float val = acc[mm][j][i] + bv;
        long oidx = (tensor < 3)
                      ? (((long)b * F_ + f) * T_ + t) * C_ + n    // [B][F][T][C]
                      : (((long)b * T_ + t) * F_ + f) * C_ + n;   // [B][T][F][C]
        outp[oidx] = val;
      }
    }
  }
}

// ---------------------------------------------------------------------------
// Freq attention: per (b,f) slice, L=T=249, d=256. One block per 16-row q-tile.
// ---------------------------------------------------------------------------
__global__ __launch_bounds__(128)
void k_attn_freq(float* __restrict__ ws, const float* __restrict__ alpha) {
  __shared__ float S[16][260];
  __shared__ float red[16][8];
  __shared__ float rs[16];

  int id = blockIdx.x;
  int qtile = id % 16; id /= 16;
  int f = id % F_;
  int b = id / F_;

  long sbase = ((long)b * F_ + f) * T_ * C_;
  const float* q  = ws + OFF_QF + sbase;
  const float* kM = ws + OFF_KF + sbase;
  const float* vM = ws + OFF_VF + sbase;
  const float* zp = ws + OFF_ZERO;
  float* gf = ws + OFF_GF;

  int wave = threadIdx.x >> 5, lane = threadIdx.x & 31;
  int l15 = lane & 15, hi = lane >> 4;
  int r0 = qtile * 16;

  // ---- Phase 1: S = q @ k^T (zero-page for OOB rows/keys; no branches) ----
  {
    v8f acc[4];
    #pragma unroll
    for (int j = 0; j < 4; ++j) acc[j] = (v8f){0.f,0.f,0.f,0.f,0.f,0.f,0.f,0.f};
    int arow = r0 + l15;
    const float* aPtr = (arow < T_) ? q + (long)arow * C_ : zp;
    const float* bPtr[4];
    #pragma unroll
    for (int j = 0; j < 4; ++j) {
      int n = (wave * 4 + j) * 16 + l15;
      bPtr[j] = (n < T_) ? kM + (long)n * C_ : zp;
    }
    for (int c0 = 0; c0 < C_; c0 += 4) {
      v2f a = *(const v2f*)(aPtr + c0 + hi * 2);
      #pragma unroll
      for (int j = 0; j < 4; ++j) {
        v2f bb = *(const v2f*)(bPtr[j] + c0 + hi * 2);
        acc[j] = wmma4(a, bb, acc[j]);
      }
    }
    #pragma unroll
    for (int j = 0; j < 4; ++j) {
      int n = (wave * 4 + j) * 16 + l15;
      #pragma unroll
      for (int i = 0; i < 8; ++i)
        S[i + hi * 8][n] = (n < T_) ? acc[j][i] : -1e30f;
    }
  }
  __syncthreads();

  // ---- Phase 2: fp32 softmax (P unnormalized; rs = row sums) ----
  {
    int row = threadIdx.x >> 3;
    int part = threadIdx.x & 7;
    float mx = -1e30f;
    for (int c = part * 32; c < part * 32 + 32; ++c) mx = fmaxf(mx, S[row][c]);
    red[row][part] = mx;
    __syncthreads();
    mx = red[row][0];
    #pragma unroll
    for (int i2 = 1; i2 < 8; ++i2) mx = fmaxf(mx, red[row][i2]);
    float sum = 0.f;
    for (int c = part * 32; c < part * 32 + 32; ++c) {
      float e = __expf(S[row][c] - mx);
      S[row][c] = e;
      sum += e;
    }
    __syncthreads();
    red[row][part] = sum;
    __syncthreads();
    if (part == 0) {
      float s2 = 0.f;
      #pragma unroll
      for (int i2 = 0; i2 < 8; ++i2) s2 += red[row][i2];
      rs[row] = s2;
    }
  }
  __syncthreads();

  // ---- Phase 3: out = P @ V, alpha-gated store to [B][T][F][C] ----
  {
    v8f acc[4];
    #pragma unroll
    for (int j = 0; j < 4; ++j) acc[j] = (v8f){0.f,0.f,0.f,0.f,0.f,0.f,0.f,0.f};
    for (int k0 = 0; k0 < C_; k0 += 4) {   // padded keys carry P == 0
      int key = k0 + hi * 2;
      v2f a = *(const v2f*)(&S[l15][key]);
      int kk0 = (key     < T_) ? key     : T_ - 1;
      int kk1 = (key + 1 < T_) ? key + 1 : T_ - 1;
      #pragma unroll
      for (int j = 0; j < 4; ++j) {
        int n = (wave * 4 + j) * 16 + l15;
        v2f bb = { vM[(long)kk0 * C_ + n], vM[(long)kk1 * C_ + n] };
        acc[j] = wmma4(a, bb, acc[j]);
      }
    }
    #pragma unroll
    for (int j = 0; j < 4; ++j) {
      int n = (wave * 4 + j) * 16 + l15;
      #pragma unroll
      for (int i = 0; i < 8; ++i) {
        int m = i + hi * 8;
        int trow = r0 + m;
        if (trow < T_) {
          float val = acc[j][i] / rs[m];
          long oidx = (((long)b * T_ + trow) * F_ + f) * C_ + n;
          gf[oidx] = alpha[oidx] * val;
        }
      }
    }
  }
}

// ---------------------------------------------------------------------------
// Time attention: per (b,t) slice, L=F=64, d=256. One block per slice.
// ---------------------------------------------------------------------------
__global__ __launch_bounds__(256)
void k_attn_time(float* __restrict__ ws, const float* __restrict__ beta) {
  __shared__ float S[64][68];
  __shared__ float red[64][4];
  __shared__ float rs[64];

  int t = blockIdx.x % T_;
  int b = blockIdx.x / T_;
  long base = ((long)b * T_ + t) * F_ * C_;
  const float* q  = ws + OFF_QT + base;
  const float* kM = ws + OFF_KT + base;
  const float* vM = ws + OFF_VT + base;
  float* gt = ws + OFF_GT;

  int wave = threadIdx.x >> 5, lane = threadIdx.x & 31;
  int l15 = lane & 15, hi = lane >> 4;

  for (int tt2 = wave * 2; tt2 < wave * 2 + 2; ++tt2) {
    int m0 = (tt2 >> 2) * 16, n0 = (tt2 & 3) * 16;
    v8f acc = (v8f){0.f,0.f,0.f,0.f,0.f,0.f,0.f,0.f};
    for (int c0 = 0; c0 < C_; c0 += 4) {
      v2f a  = *(const v2f*)(q  + (long)(m0 + l15) * C_ + c0 + hi * 2);
      v2f bb = *(const v2f*)(kM + (long)(n0 + l15) * C_ + c0 + hi * 2);
      acc = wmma4(a, bb, acc);
    }
    #pragma unroll
    for (int i = 0; i < 8; ++i) S[m0 + i + hi * 8][n0 + l15] = acc[i];
  }
  __syncthreads();

  {
    int row = threadIdx.x >> 2, part = threadIdx.x & 3;
    float mx = -1e30f;
    for (int c = part * 16; c < part * 16 + 16; ++c) mx = fmaxf(mx, S[row][c]);
    red[row][part] = mx;
    __syncthreads();
    mx = fmaxf(fmaxf(red[row][0], red[row][1]), fmaxf(red[row][2], red[row][3]));
    float sum = 0.f;
    for (int c = part * 16; c < part * 16 + 16; ++c) {
      float e = __expf(S[row][c] - mx);
      S[row][c] = e;
      sum += e;
    }
    __syncthreads();
    red[row][part] = sum;
    __syncthreads();
    if (part == 0) rs[row] = red[row][0] + red[row][1] + red[row][2] + red[row][3];
  }
  __syncthreads();

  for (int tt2 = wave * 8; tt2 < wave * 8 + 8; ++tt2) {
    int m0 = (tt2 >> 4) * 16, n0 = (tt2 & 15) * 16;
    v8f acc = (v8f){0.f,0.f,0.f,0.f,0.f,0.f,0.f,0.f};
    for (int k0 = 0; k0 < F_; k0 += 4) {
      int key = k0 + hi * 2;
      v2f a = *(const v2f*)(&S[m0 + l15][key]);
      v2f bb = { vM[(long)key * C_ + n0 + l15],
                 vM[(long)(key + 1) * C_ + n0 + l15] };
      acc = wmma4(a, bb, acc);
    }
    #pragma unroll
    for (int i = 0; i < 8; ++i) {
      int fr = m0 + i + hi * 8;
      float val = acc[i] / rs[fr];
      long oidx = base + (long)fr * C_ + (n0 + l15);
      gt[oidx] = beta[oidx] * val;
    }
  }
}

// ---------------------------------------------------------------------------
// Final conv: implicit concat [gf | gt | x], K=6912, N=64, M=32 per block.
// ---------------------------------------------------------------------------
__global__ __launch_bounds__(128)
void k_final_conv(const float* __restrict__ x, float* __restrict__ ws,
                  const float* __restrict__ bfin, float* __restrict__ out) {
  int id = blockIdx.x;
  int half = id % 2; id /= 2;
  int t = id % T_;
  int b = id / T_;
  int wave = threadIdx.x >> 5, lane = threadIdx.x & 31;
  int l15 = lane & 15, hi = lane >> 4;
  int n0 = wave * 16;
  int f0 = half * 32;

  const float* wfin = ws + OFF_WFIN;
  const float* gf = ws + OFF_GF;
  const float* gt = ws + OFF_GT;
  const float* zp = ws + OFF_ZERO;

  v8f acc[2];
  acc[0] = (v8f){0.f,0.f,0.f,0.f,0.f,0.f,0.f,0.f};
  acc[1] = (v8f){0.f,0.f,0.f,0.f,0.f,0.f,0.f,0.f};

  const float* wRow = wfin + (long)(n0 + l15) * KFIN;

  for (int p = 0; p < 9; ++p) {
    int dt = p / 3 - 1, df = p % 3 - 1;
    int tt = t + dt;
    bool tOK = (unsigned)tt < (unsigned)T_;
    long sidx[2];
    bool mv[2];
    #pragma unroll
    for (int mm = 0; mm < 2; ++mm) {
      int ff = f0 + mm * 16 + l15 + df;
      mv[mm] = tOK && ((unsigned)ff < (unsigned)F_);
      sidx[mm] = ((long)b * T_ + tt) * F_ + ff;
    }
    for (int grp = 0; grp < 3; ++grp) {
      const float* src = (grp == 0) ? gf : (grp == 1) ? gt : x;
      const float* aP0 = mv[0] ? src + sidx[0] * C_ : zp;
      const float* aP1 = mv[1] ? src + sidx[1] * C_ : zp;
      int kbase = p * 768 + grp * C_;
      for (int c0 = 0; c0 < C_; c0 += 4) {
        v2f a0 = *(const v2f*)(aP0 + c0 + hi * 2);
        v2f a1 = *(const v2f*)(aP1 + c0 + hi * 2);
        v2f bb = *(const v2f*)(wRow + kbase + c0 + hi * 2);
        acc[0] = wmma4(a0, bb, acc[0]);
        acc[1] = wmma4(a1, bb, acc[1]);
      }
    }
  }
  int o = n0 + l15;
  float bv = bfin[o];
  #pragma unroll
  for (int mm = 0; mm < 2; ++mm) {
    #pragma unroll
    for (int i = 0; i < 8; ++i) {
      int fr = f0 + mm * 16 + i + hi * 8;
      out[(((long)b * T_ + t) * F_ + fr) * 64 + o] = acc[mm][i] + bv;
    }
  }
}

// ---------------------------------------------------------------------------
// Host launch.
// ---------------------------------------------------------------------------
extern "C" void kernel_launch(void* const* d_in, const int* in_sizes, int n_in,
                              void* d_out, int out_size, void* d_ws, size_t ws_size,
                              hipStream_t stream) {
  (void)in_sizes; (void)n_in; (void)out_size; (void)ws_size;

  const float* x    = (const float*)d_in[0];
  const float* wqf  = (const float*)d_in[1];
  const float* bqf  = (const float*)d_in[2];
  const float* wkf  = (const float*)d_in[3];
  const float* bkf  = (const float*)d_in[4];
  const float* wvf  = (const float*)d_in[5];
  const float* bvf  = (const float*)d_in[6];
  const float* wqt  = (const float*)d_in[7];
  const float* bqt  = (const float*)d_in[8];
  const float* wkt  = (const float*)d_in[9];
  const float* bkt  = (const float*)d_in[10];
  const float* wvt  = (const float*)d_in[11];
  const float* bvt  = (const float*)d_in[12];
  const float* wfin = (const float*)d_in[13];
  const float* bfin = (const float*)d_in[14];
  const float* alpha = (const float*)d_in[15];
  const float* beta  = (const float*)d_in[16];
  float* out = (float*)d_out;
  float* ws  = (float*)d_ws;

  k_zero<<<1, 256, 0, stream>>>(ws + OFF_ZERO, 256);
  {
    long tot = (long)KQK * C_;
    int thr = 256, blk = (int)((tot + thr - 1) / thr);
    k_freq_repack<<<blk, thr, 0, stream>>>(wqf, ws + OFF_WB + 0 * WSZ_QKV);
    k_freq_repack<<<blk, thr, 0, stream>>>(wkf, ws + OFF_WB + 1 * WSZ_QKV);
    k_freq_repack<<<blk, thr, 0, stream>>>(wvf, ws + OFF_WB + 2 * WSZ_QKV);
    k_transpose <<<blk, thr, 0, stream>>>(wqt, ws + OFF_WB + 3 * WSZ_QKV, KQK, C_);
    k_transpose <<<blk, thr, 0, stream>>>(wkt, ws + OFF_WB + 4 * WSZ_QKV, KQK, C_);
    k_transpose <<<blk, thr, 0, stream>>>(wvt, ws + OFF_WB + 5 * WSZ_QKV, KQK, C_);
    long totF = (long)KFIN * 64;
    k_transpose<<<(int)((totF + 255) / 256), 256, 0, stream>>>(
        wfin, ws + OFF_WFIN, KFIN, 64);
  }

  Bias6 bp;
  bp.p[0] = bqf; bp.p[1] = bkf; bp.p[2] = bvf;
  bp.p[3] = bqt; bp.p[4] = bkt; bp.p[5] = bvt;

  k_qkv_conv  <<<B_ * T_ * 2 * 6, 128, 0, stream>>>(x, ws, bp);
  k_attn_freq <<<B_ * F_ * 16,    128, 0, stream>>>(ws, alpha);
  k_attn_time <<<B_ * T_,         256, 0, stream>>>(ws, beta);
  k_final_conv<<<B_ * T_ * 2,     128, 0, stream>>>(x, ws, bfin, out);
}